// LengthRegulator_42691974922415
// MI455X (gfx1250) — compile-verified
//
#include <hip/hip_runtime.h>
#include <stdint.h>

// LengthRegulator for MI455X (gfx1250).
// B=32, L=512, D=384, DUR_MAX=8 -> T = L*(DUR_MAX-1) = 3584 output frames/batch.
// Memory-bound (~201 MB HBM traffic, ~8.6 us floor @ 23.3 TB/s). No matmul ->
// no WMMA; use CDNA5 data-movement paths instead: async global<->LDS copies
// (ASYNCcnt-tracked) for valid rows, direct NT b128 stores for padding rows.

#define B_  32
#define L_  512
#define D_  384
#define T_  3584   // L_*(DUR_MAX-1)
#define D4_ 96     // D_/4 (float4 chunks per row)

typedef float v4f __attribute__((ext_vector_type(4)));
typedef int   v4i __attribute__((vector_size(16)));   // matches builtin param type

#define AS1 __attribute__((address_space(1)))
#define AS3 __attribute__((address_space(3)))

// Async global<->LDS builtins exist only on newer gfx1250 toolchains; probe.
#if defined(__HIP_DEVICE_COMPILE__) && defined(__gfx1250__) && \
    __has_builtin(__builtin_amdgcn_global_load_async_to_lds_b128) && \
    __has_builtin(__builtin_amdgcn_global_store_async_from_lds_b128)
#define LR_ASYNC 1
#else
#define LR_ASYNC 0
#endif

#if defined(__HIP_DEVICE_COMPILE__) && defined(__gfx1250__)
#if __has_builtin(__builtin_amdgcn_s_wait_asynccnt)
#define LR_WAIT_ASYNC0() __builtin_amdgcn_s_wait_asynccnt(0)
#else
#define LR_WAIT_ASYNC0() asm volatile("s_wait_asynccnt 0" ::: "memory")
#endif
#else
#define LR_WAIT_ASYNC0()
#endif

// ---------------------------------------------------------------------------
// Kernel: expand. One wave (32 lanes) per output row; 4 rows per 128-thread
// block. Each row = 1536 B = 3 x b128 per lane. No block-wide sync needed:
// each wave owns its private LDS staging slot.
// ---------------------------------------------------------------------------
__global__ __launch_bounds__(128) void lr_expand(const float* __restrict__ hid,
                                                 const int* __restrict__ idxmap,
                                                 float* __restrict__ out) {
  const int tid  = threadIdx.x;
  const int wave = tid >> 5;
  const int lane = tid & 31;
  const int row  = blockIdx.x * 4 + wave;
  const int b    = row / T_;
  const int idx  = idxmap[row];

#if LR_ASYNC
  __shared__ float slot[4][D_];    // one staging slot per wave
  float* orow = out + (size_t)row * D_;
  if (idx >= 0) {
    // Valid frame: ASYNCcnt-tracked global -> LDS -> global copy; the wave's
    // VGPRs never touch the payload.
    const float* srow = hid + ((size_t)b * L_ + idx) * D_;
    float* my = slot[wave];
    #pragma unroll
    for (int p = 0; p < 3; ++p) {
      __builtin_amdgcn_global_load_async_to_lds_b128(
          (AS1 v4i*)(uintptr_t)(srow + (lane + 32 * p) * 4),
          (AS3 v4i*)(uint32_t)(uintptr_t)(my + (lane + 32 * p) * 4),
          0, 0);
    }
    LR_WAIT_ASYNC0();              // async loads fully complete -> LDS valid
    #pragma unroll
    for (int p = 0; p < 3; ++p) {
      __builtin_amdgcn_global_store_async_from_lds_b128(
          (AS1 v4i*)(uintptr_t)(orow + (lane + 32 * p) * 4),
          (AS3 v4i*)(uint32_t)(uintptr_t)(my + (lane + 32 * p) * 4),
          0, 0);
    }
    // S_ENDPGM performs an implicit wait-idle; stores drain before wave exit.
  } else {
    // Padding frame (~50% of rows): direct zero fill, non-temporal (write-once
    // stream, keep L2 for the reused hidden rows).
    v4f z = {0.0f, 0.0f, 0.0f, 0.0f};
    v4f* o4 = (v4f*)orow;
    __builtin_nontemporal_store(z, o4 + lane);
    __builtin_nontemporal_store(z, o4 + lane + 32);
    __builtin_nontemporal_store(z, o4 + lane + 64);
  }
#else
  // Fallback: direct b128 copy, NT stores + global prefetch of the source row.
  v4f* orow = (v4f*)out + (size_t)row * D4_;
  if (idx >= 0) {
    const v4f* srow = (const v4f*)hid + ((size_t)b * L_ + idx) * D4_;
    __builtin_prefetch(srow + lane, 0, 1);   // -> global_prefetch_b8
    v4f a0 = srow[lane];
    v4f a1 = srow[lane + 32];
    v4f a2 = srow[lane + 64];
    __builtin_nontemporal_store(a0, orow + lane);
    __builtin_nontemporal_store(a1, orow + lane + 32);
    __builtin_nontemporal_store(a2, orow + lane + 64);
  } else {
    v4f z = {0.0f, 0.0f, 0.0f, 0.0f};
    __builtin_nontemporal_store(z, orow + lane);
    __builtin_nontemporal_store(z, orow + lane + 32);
    __builtin_nontemporal_store(z, orow + lane + 64);
  }
#endif
}

// ---------------------------------------------------------------------------
// Kernel: per-batch inclusive scan of durations + scatter frame->phoneme map.
// One 512-thread block per batch. idxmap[b*T + t] = phoneme index, or -1 (pad).
// ---------------------------------------------------------------------------
__global__ __launch_bounds__(L_) void lr_build_index(const int* __restrict__ dur,
                                                     int* __restrict__ idxmap) {
  __shared__ int cum[L_];
  const int b   = blockIdx.x;
  const int tid = threadIdx.x;

  // Init all frames of this batch to "padding".
  for (int t = tid; t < T_; t += L_) idxmap[b * T_ + t] = -1;

  const int d = dur[b * L_ + tid];
  cum[tid] = d;
  __syncthreads();

  // Hillis-Steele inclusive scan over 512 elements.
  #pragma unroll
  for (int off = 1; off < L_; off <<= 1) {
    int v = (tid >= off) ? cum[tid - off] : 0;
    __syncthreads();
    cum[tid] += v;
    __syncthreads();
  }

  // Scatter: phoneme tid owns frames [cum-d, cum). Write-disjoint; cum<=T_.
  const int end = cum[tid];
  for (int k = end - d; k < end; ++k) idxmap[b * T_ + k] = tid;
}

// ---------------------------------------------------------------------------
// Host launcher.
// d_in[0]: pred_duration  int32 [B, L]
// d_in[1]: phoneme_hidden float32 [B, L, D]
// d_out  : float32 [B, T, D]
// d_ws   : >= B*T*4 bytes (frame -> phoneme index map)
// ---------------------------------------------------------------------------
extern "C" void kernel_launch(void* const* d_in, const int* in_sizes, int n_in,
                              void* d_out, int out_size, void* d_ws, size_t ws_size,
                              hipStream_t stream) {
  (void)in_sizes; (void)n_in; (void)out_size; (void)ws_size;
  const int*   dur = (const int*)d_in[0];
  const float* hid = (const float*)d_in[1];
  float*       out = (float*)d_out;
  int*      idxmap = (int*)d_ws;

  lr_build_index<<<B_, L_, 0, stream>>>(dur, idxmap);
  lr_expand<<<(B_ * T_) / 4, 128, 0, stream>>>(hid, idxmap, out);
}